// GraniteMoeParallelExperts_27865747816885
// MI455X (gfx1250) — compile-verified
//
#include <hip/hip_runtime.h>

// MoE grouped GEMM: Y[e] = X[e] @ W[e]^T
// X: [64, 4096, 1024] f32, W: [64, 512, 1024] f32, Y: [64, 4096, 512] f32
// HBM-bound (~1.74 GB @ 23.3 TB/s ~= 75us); f32 WMMA keeps full precision for free.
// Double-buffered LDS: one barrier per K-tile, global loads for tile t+1 in
// flight while tile t runs through v_wmma_f32_16x16x4_f32.

typedef __attribute__((ext_vector_type(2))) float v2f;
typedef __attribute__((ext_vector_type(4))) float v4f;
typedef __attribute__((ext_vector_type(8))) float v8f;

#define NUM_EXPERTS 64
#define K_SIZE      1024
#define N_SIZE      512
#define M_PER_EXP   4096

#define BM 128
#define BN 64
#define BK 32
#define LDSTR (BK + 4)   // 36 floats: 16B-aligned rows, conflict-free fragment reads

__global__ __launch_bounds__(256) void moe_gemm_wmma_f32(
    const float* __restrict__ X, const float* __restrict__ W, float* __restrict__ Y)
{
    __shared__ float lds_a[2][BM * LDSTR];   // 2 x 18 KB
    __shared__ float lds_b[2][BN * LDSTR];   // 2 x  9 KB

    const int tid  = threadIdx.x;
    const int lane = tid & 31;
    const int wave = tid >> 5;

    const int e  = blockIdx.z;
    const int m0 = blockIdx.y * BM;
    const int n0 = blockIdx.x * BN;

    const float* Xe = X + (size_t)e * M_PER_EXP * K_SIZE + (size_t)m0 * K_SIZE;
    const float* We = W + (size_t)e * N_SIZE   * K_SIZE + (size_t)n0 * K_SIZE;

    // 8 waves: 4 along M, 2 along N; each wave computes 32x32 via 2x2 WMMA tiles
    const int wm = (wave & 3) << 5;   // 0,32,64,96
    const int wn = (wave >> 2) << 5;  // 0,32

    const int lane15   = lane & 15;
    const int kLaneOff = (lane & 16) >> 3;   // 0 for lanes 0-15, 2 for lanes 16-31
    const int rowAdd   = (lane & 16) >> 1;   // 0 / 8 for C/D epilogue

    v8f acc[2][2];
    #pragma unroll
    for (int i = 0; i < 2; ++i)
        #pragma unroll
        for (int j = 0; j < 2; ++j)
            acc[i][j] = (v8f)(0.0f);

    // Cooperative tile-load indexing: 8 lanes/row, b128 each -> 32 rows per pass
    const int lrow = tid >> 3;          // 0..31
    const int lcol = (tid & 7) << 2;    // 0,4,...,28

    v4f regA[4], regB[2];

    // ---- prologue: stage tile 0 into buffer 0 ----
    #pragma unroll
    for (int p = 0; p < 4; ++p)
        regA[p] = *(const v4f*)(Xe + (size_t)((p << 5) + lrow) * K_SIZE + lcol);
    #pragma unroll
    for (int p = 0; p < 2; ++p)
        regB[p] = *(const v4f*)(We + (size_t)((p << 5) + lrow) * K_SIZE + lcol);
    #pragma unroll
    for (int p = 0; p < 4; ++p)
        *(v4f*)(&lds_a[0][((p << 5) + lrow) * LDSTR + lcol]) = regA[p];
    #pragma unroll
    for (int p = 0; p < 2; ++p)
        *(v4f*)(&lds_b[0][((p << 5) + lrow) * LDSTR + lcol]) = regB[p];

    int cur = 0;
    for (int kt = 0; kt < K_SIZE; kt += BK) {
        __syncthreads();   // buf[cur] stores visible; buf[cur^1] reads (from t-1) done

        const bool more = (kt + BK) < K_SIZE;
        if (more) {
            // issue next tile's global loads now; they complete under the compute
            #pragma unroll
            for (int p = 0; p < 4; ++p)
                regA[p] = *(const v4f*)(Xe + (size_t)((p << 5) + lrow) * K_SIZE + (kt + BK) + lcol);
            #pragma unroll
            for (int p = 0; p < 2; ++p)
                regB[p] = *(const v4f*)(We + (size_t)((p << 5) + lrow) * K_SIZE + (kt + BK) + lcol);
        }

        // ---- compute: 8 k-steps of V_WMMA_F32_16X16X4_F32 on buf[cur] ----
        const float* __restrict__ la = &lds_a[cur][0];
        const float* __restrict__ lb = &lds_b[cur][0];
        #pragma unroll
        for (int kk = 0; kk < BK; kk += 4) {
            v2f a[2], b[2];
            #pragma unroll
            for (int f = 0; f < 2; ++f)
                a[f] = *(const v2f*)(&la[(wm + (f << 4) + lane15) * LDSTR + kk + kLaneOff]);
            #pragma unroll
            for (int f = 0; f < 2; ++f)
                b[f] = *(const v2f*)(&lb[(wn + (f << 4) + lane15) * LDSTR + kk + kLaneOff]);

            #pragma unroll
            for (int i = 0; i < 2; ++i)
                #pragma unroll
                for (int j = 0; j < 2; ++j)
                    acc[i][j] = __builtin_amdgcn_wmma_f32_16x16x4_f32(
                        /*neg_a=*/false, a[i], /*neg_b=*/false, b[j],
                        /*c_mod=*/(short)0, acc[i][j],
                        /*reuse_a=*/false, /*reuse_b=*/false);
        }

        if (more) {
            // stage next tile into the other buffer (reads of it finished before
            // this iteration's barrier)
            const int nxt = cur ^ 1;
            #pragma unroll
            for (int p = 0; p < 4; ++p)
                *(v4f*)(&lds_a[nxt][((p << 5) + lrow) * LDSTR + lcol]) = regA[p];
            #pragma unroll
            for (int p = 0; p < 2; ++p)
                *(v4f*)(&lds_b[nxt][((p << 5) + lrow) * LDSTR + lcol]) = regB[p];
        }
        cur ^= 1;
    }

    // ---- epilogue: C/D layout -> global. VGPR v: lanes0-15 M=v, lanes16-31 M=v+8; N=lane15
    float* Ye = Y + (size_t)e * M_PER_EXP * N_SIZE;
    #pragma unroll
    for (int i = 0; i < 2; ++i) {
        #pragma unroll
        for (int j = 0; j < 2; ++j) {
            const int col = n0 + wn + (j << 4) + lane15;
            #pragma unroll
            for (int v = 0; v < 8; ++v) {
                const int row = m0 + wm + (i << 4) + rowAdd + v;
                Ye[(size_t)row * N_SIZE + col] = acc[i][j][v];
            }
        }
    }
}

extern "C" void kernel_launch(void* const* d_in, const int* in_sizes, int n_in,
                              void* d_out, int out_size, void* d_ws, size_t ws_size,
                              hipStream_t stream) {
    const float* X = (const float*)d_in[0];
    const float* W = (const float*)d_in[1];
    float* Y = (float*)d_out;

    dim3 grid(N_SIZE / BN, M_PER_EXP / BM, NUM_EXPERTS);  // (8, 32, 64)
    dim3 block(256);
    moe_gemm_wmma_f32<<<grid, block, 0, stream>>>(X, W, Y);
}